// SelectiveSSM_12936441495989
// MI455X (gfx1250) — compile-verified
//
#include <hip/hip_runtime.h>
#include <math.h>

typedef __attribute__((ext_vector_type(2))) float v2f;
typedef __attribute__((ext_vector_type(8))) float v8f;

// Problem constants
#define BATCH   2
#define SEQ     512
#define DMODEL  768
#define DINNER  1536
#define DSTATE  16
#define DCONV   4
#define MROWS   (BATCH * SEQ)          // 1024
#define DXZ     (2 * DINNER)           // 3072
#define DXDBL   (DINNER + 2 * DSTATE)  // 1568

__device__ __forceinline__ float softplusf(float x) {
    // log(1 + exp(x)) computed stably, matches jax.nn.softplus
    return fmaxf(x, 0.f) + log1pf(expf(-fabsf(x)));
}
__device__ __forceinline__ float siluf(float x) {
    return x / (1.f + expf(-x));
}

#define WMMA_F32(ACC, AV, BV) \
    ACC = __builtin_amdgcn_wmma_f32_16x16x4_f32(false, AV, false, BV, \
                                                (short)0, ACC, false, false)

// ---------------------------------------------------------------------------
// fp32 WMMA GEMM:  C[M x N] = A[M x K] @ W[N x K]^T   (+ optional bias/softplus)
// One wave computes one 16-row M tile x FOUR 16-col N tiles (4 accumulators,
// shared A fragment).  Copy-free ping-pong pipeline (depth 2): two fragment
// slot-sets written only by loads and read only by WMMAs, so ~5 loads stay in
// flight under every WMMA group.  Requires K/4 even (holds for K=768/1536).
// V_WMMA_F32_16X16X4_F32 fragment layout (wave32):
//   A (16x4): lane = M%16 in each half-wave; half 0 holds K={k,k+1}, half 1 K={k+2,k+3}
//   B (4x16): same with N in place of M
//   C/D (16x16): VGPR r, half h -> M = r + 8*h, N = lane%16
// N-tail: last block clamps n0 to N-64 (overlapping recompute keeps EXEC all-1s).
// MODE 0: plain store.  MODE 1: +bias then softplus (for the delta GEMM).
// ---------------------------------------------------------------------------
template<int MODE>
__global__ __launch_bounds__(32)
void gemm_wmma_f32(const float* __restrict__ A,    // M x K, row stride lda
                   const float* __restrict__ Wt,   // N x K, row stride K
                   const float* __restrict__ bias, // [N] or nullptr
                   float* __restrict__ C,          // M x ldc
                   int K, int lda, int ldc, int N)
{
    const int lane = threadIdx.x;
    const int half = lane >> 4;
    const int mn   = lane & 15;
    const int koff = half * 2;
    const int tileM = blockIdx.y;

    int n0 = blockIdx.x * 64;              // four adjacent 16-wide N tiles
    if (n0 + 64 > N) n0 = N - 64;          // overlapping tail (duplicate work)

    const float* ap  = A  + (size_t)(tileM * 16 + mn) * lda + koff;
    const float* bp0 = Wt + (size_t)(n0      + mn) * K + koff;
    const float* bp1 = Wt + (size_t)(n0 + 16 + mn) * K + koff;
    const float* bp2 = Wt + (size_t)(n0 + 32 + mn) * K + koff;
    const float* bp3 = Wt + (size_t)(n0 + 48 + mn) * K + koff;

    v8f acc0 = {}, acc1 = {}, acc2 = {}, acc3 = {};

    // slot A <- K-step 0, slot B <- K-step 1
    v2f aA  = *(const v2f*)(ap);
    v2f bA0 = *(const v2f*)(bp0);
    v2f bA1 = *(const v2f*)(bp1);
    v2f bA2 = *(const v2f*)(bp2);
    v2f bA3 = *(const v2f*)(bp3);
    v2f aB  = *(const v2f*)(ap  + 4);
    v2f bB0 = *(const v2f*)(bp0 + 4);
    v2f bB1 = *(const v2f*)(bp1 + 4);
    v2f bB2 = *(const v2f*)(bp2 + 4);
    v2f bB3 = *(const v2f*)(bp3 + 4);

    for (int k = 8; k < K; k += 8) {
        WMMA_F32(acc0, aA, bA0);
        WMMA_F32(acc1, aA, bA1);
        WMMA_F32(acc2, aA, bA2);
        WMMA_F32(acc3, aA, bA3);
        aA  = *(const v2f*)(ap  + k);
        bA0 = *(const v2f*)(bp0 + k);
        bA1 = *(const v2f*)(bp1 + k);
        bA2 = *(const v2f*)(bp2 + k);
        bA3 = *(const v2f*)(bp3 + k);
        WMMA_F32(acc0, aB, bB0);
        WMMA_F32(acc1, aB, bB1);
        WMMA_F32(acc2, aB, bB2);
        WMMA_F32(acc3, aB, bB3);
        aB  = *(const v2f*)(ap  + k + 4);
        bB0 = *(const v2f*)(bp0 + k + 4);
        bB1 = *(const v2f*)(bp1 + k + 4);
        bB2 = *(const v2f*)(bp2 + k + 4);
        bB3 = *(const v2f*)(bp3 + k + 4);
    }
    // drain the two live slots
    WMMA_F32(acc0, aA, bA0);
    WMMA_F32(acc1, aA, bA1);
    WMMA_F32(acc2, aA, bA2);
    WMMA_F32(acc3, aA, bA3);
    WMMA_F32(acc0, aB, bB0);
    WMMA_F32(acc1, aB, bB1);
    WMMA_F32(acc2, aB, bB2);
    WMMA_F32(acc3, aB, bB3);

    float bv0 = 0.f, bv1 = 0.f, bv2 = 0.f, bv3 = 0.f;
    if (MODE == 1) {
        bv0 = bias[n0      + mn];
        bv1 = bias[n0 + 16 + mn];
        bv2 = bias[n0 + 32 + mn];
        bv3 = bias[n0 + 48 + mn];
    }

#pragma unroll
    for (int r = 0; r < 8; ++r) {
        int m = tileM * 16 + half * 8 + r;
        float v0 = acc0[r], v1 = acc1[r], v2 = acc2[r], v3 = acc3[r];
        if (MODE == 1) {
            v0 = softplusf(v0 + bv0);
            v1 = softplusf(v1 + bv1);
            v2 = softplusf(v2 + bv2);
            v3 = softplusf(v3 + bv3);
        }
        float* crow = C + (size_t)m * ldc + n0 + mn;
        crow[0]  = v0;
        crow[16] = v1;
        crow[32] = v2;
        crow[48] = v3;
    }
}

// ---------------------------------------------------------------------------
// Depthwise causal conv (kernel 4) + bias + SiLU on x_inner = xz[:, :DINNER]
// ---------------------------------------------------------------------------
__global__ __launch_bounds__(256)
void conv_silu_kernel(const float* __restrict__ xz,
                      const float* __restrict__ cw,   // (DINNER, 1, 4)
                      const float* __restrict__ cb,   // (DINNER,)
                      float* __restrict__ xconv)      // (MROWS, DINNER)
{
    int idx = blockIdx.x * 256 + threadIdx.x;
    const int total = MROWS * DINNER;
    if (idx >= total) return;
    int d   = idx % DINNER;
    int row = idx / DINNER;         // b*SEQ + l
    int l   = row & (SEQ - 1);
    float acc = cb[d];
#pragma unroll
    for (int j = 0; j < DCONV; ++j) {
        int li = l - (DCONV - 1) + j;
        if (li >= 0)
            acc = fmaf(cw[d * DCONV + j],
                       xz[(size_t)(row - (DCONV - 1) + j) * DXZ + d], acc);
    }
    xconv[idx] = siluf(acc);
}

// ---------------------------------------------------------------------------
// Selective scan, one lane per (b, d, n).  Faithful to the log-space scan:
//   c[t] = exp(delta*A) + 1e-12 ;  v[t] = |delta*B*x_conv| + 1e-12
//   h[t] = c[t]*h[t-1] + v[t]
// 16-lane shuffle reduction contracts over n with C; fuses D-skip + SiLU(z).
// ---------------------------------------------------------------------------
__global__ __launch_bounds__(256)
void ssm_scan_kernel(const float* __restrict__ delta,  // (MROWS, DINNER)
                     const float* __restrict__ xdbl,   // (MROWS, DXDBL)
                     const float* __restrict__ xconv,  // (MROWS, DINNER)
                     const float* __restrict__ xz,     // (MROWS, DXZ) for z
                     const float* __restrict__ A_log,  // (DINNER, DSTATE)
                     const float* __restrict__ Dp,     // (DINNER,)
                     float* __restrict__ y)            // (MROWS, DINNER)
{
    int tid = blockIdx.x * 256 + threadIdx.x;          // 0 .. BATCH*DINNER*16-1
    int n  = tid & 15;
    int bd = tid >> 4;
    if (bd >= BATCH * DINNER) return;
    int d = bd % DINNER;
    int b = bd / DINNER;

    float An = -expf(A_log[d * DSTATE + n]);
    float Dd = Dp[d];
    float h  = 0.f;

    for (int t = 0; t < SEQ; ++t) {
        int row = b * SEQ + t;
        float dv = delta[(size_t)row * DINNER + d];
        float xc = xconv[(size_t)row * DINNER + d];
        const float* bc = xdbl + (size_t)row * DXDBL + DINNER;
        float Bn = bc[n];
        float Cn = bc[DSTATE + n];

        float c = expf(dv * An) + 1e-12f;
        float v = fabsf(dv * Bn * xc) + 1e-12f;
        h = fmaf(c, h, v);

        float contrib = h * Cn;
        contrib += __shfl_xor(contrib, 1, 32);
        contrib += __shfl_xor(contrib, 2, 32);
        contrib += __shfl_xor(contrib, 4, 32);
        contrib += __shfl_xor(contrib, 8, 32);

        if (n == 0) {
            float zv = xz[(size_t)row * DXZ + DINNER + d];
            float yv = contrib + xc * Dd;
            yv *= siluf(zv);
            y[(size_t)row * DINNER + d] = yv;
        }
    }
}

extern "C" void kernel_launch(void* const* d_in, const int* in_sizes, int n_in,
                              void* d_out, int out_size, void* d_ws, size_t ws_size,
                              hipStream_t stream) {
    const float* x      = (const float*)d_in[0];
    const float* W_in   = (const float*)d_in[1];
    const float* conv_w = (const float*)d_in[2];
    const float* conv_b = (const float*)d_in[3];
    const float* W_x    = (const float*)d_in[4];
    const float* W_dt   = (const float*)d_in[5];
    const float* b_dt   = (const float*)d_in[6];
    const float* A_log  = (const float*)d_in[7];
    const float* Dp     = (const float*)d_in[8];
    const float* W_out  = (const float*)d_in[9];
    float* out = (float*)d_out;

    // Workspace layout (floats): ~38 MB total
    float* ws    = (float*)d_ws;
    float* xz    = ws;                                    // MROWS x DXZ
    float* xconv = xz    + (size_t)MROWS * DXZ;           // MROWS x DINNER
    float* xdbl  = xconv + (size_t)MROWS * DINNER;        // MROWS x DXDBL
    float* delta = xdbl  + (size_t)MROWS * DXDBL;         // MROWS x DINNER
    float* yb    = delta + (size_t)MROWS * DINNER;        // MROWS x DINNER

    const dim3 wave(32, 1, 1);
    const int mTiles = MROWS / 16;                        // 64

    // 1) xz = x @ W_in^T          (K=768,  N=3072)
    gemm_wmma_f32<0><<<dim3((DXZ + 63) / 64, mTiles), wave, 0, stream>>>(
        x, W_in, nullptr, xz, DMODEL, DMODEL, DXZ, DXZ);

    // 2) depthwise conv + SiLU -> x_conv
    conv_silu_kernel<<<(MROWS * DINNER + 255) / 256, 256, 0, stream>>>(
        xz, conv_w, conv_b, xconv);

    // 3) x_dbl = x_conv @ W_x^T   (K=1536, N=1568; tail block overlaps)
    gemm_wmma_f32<0><<<dim3((DXDBL + 63) / 64, mTiles), wave, 0, stream>>>(
        xconv, W_x, nullptr, xdbl, DINNER, DINNER, DXDBL, DXDBL);

    // 4) delta = softplus(x_dbl[:, :1536] @ W_dt^T + b_dt)
    gemm_wmma_f32<1><<<dim3((DINNER + 63) / 64, mTiles), wave, 0, stream>>>(
        xdbl, W_dt, b_dt, delta, DINNER, DXDBL, DINNER, DINNER);

    // 5) selective scan + C contraction + D skip + SiLU(z) gate
    ssm_scan_kernel<<<(BATCH * DINNER * DSTATE + 255) / 256, 256, 0, stream>>>(
        delta, xdbl, xconv, xz, A_log, Dp, yb);

    // 6) out = y @ W_out^T        (K=1536, N=768)
    gemm_wmma_f32<0><<<dim3((DMODEL + 63) / 64, mTiles), wave, 0, stream>>>(
        yb, W_out, nullptr, out, DINNER, DINNER, DMODEL, DMODEL);
}